// GCNModel_52072183497149
// MI455X (gfx1250) — compile-verified
//
#include <hip/hip_runtime.h>
#include <hip/hip_bf16.h>

// GCN (3x GCNConv + mean-pool + 2-layer MLP) for MI455X (gfx1250, wave32).
//
// Strategy:
//  * Edge aggregation (the dominant cost) = one wave per edge, float4 gather of
//    hw[src] (contiguous 512B per wave) + 4 global f32 atomics per lane into
//    agg[dst]. h/hw/agg are 51.2MB each -> L2 (192MB) resident, so the random
//    phase runs out of L2, not HBM.
//  * Dense GEMMs (h@W 100000x128x128, pooled@fc1 512x128x64) use
//    V_WMMA_F32_16X16X4_F32 (full f32 precision, one wave per 16x16 tile).
//  * Layer 0 is [N,1]@[1,128] -> scalar outer product.
//  * fc2 is a 64-dot per graph -> wave32 shuffle reduction.

typedef __attribute__((ext_vector_type(2))) float v2f;
typedef __attribute__((ext_vector_type(8))) float v8f;

__global__ void fill_f32(float* __restrict__ p, float v, int n) {
    int i = blockIdx.x * blockDim.x + threadIdx.x;
    if (i < n) p[i] = v;
}

// deg[dst[e]] += 1  (deg pre-filled with 1.0 for the self loop)
__global__ void degree_kernel(const int* __restrict__ dst, float* __restrict__ deg, int E) {
    int e = blockIdx.x * blockDim.x + threadIdx.x;
    if (e < E) atomicAdd(&deg[dst[e]], 1.0f);
}

__global__ void rsqrt_kernel(const float* __restrict__ deg, float* __restrict__ dinv, int n) {
    int i = blockIdx.x * blockDim.x + threadIdx.x;
    if (i < n) dinv[i] = rsqrtf(deg[i]);   // deg >= 1 always (self loops)
}

// hw[i][j] = x[i] * W0[j]   (layer-0 "GEMM" is an outer product, Din=1)
__global__ void outer_kernel(const float* __restrict__ x, const float* __restrict__ W0,
                             float* __restrict__ hw, int n4) {  // n4 = N * (H/4)
    int t = blockIdx.x * blockDim.x + threadIdx.x;
    if (t >= n4) return;
    int i = t >> 5;          // node (H/4 == 32)
    int q = t & 31;          // float4 index within row
    float4 w = ((const float4*)W0)[q];
    float xv = x[i];
    float4 r;
    r.x = xv * w.x; r.y = xv * w.y; r.z = xv * w.z; r.w = xv * w.w;
    ((float4*)hw)[t] = r;
}

// One wave (32 lanes) per edge; lane handles 4 consecutive features.
__global__ void edge_agg_kernel(const int* __restrict__ src, const int* __restrict__ dst,
                                const float* __restrict__ dinv, const float* __restrict__ hw,
                                float* __restrict__ agg, int E) {
    int t = blockIdx.x * blockDim.x + threadIdx.x;
    int e = t >> 5;
    if (e >= E) return;
    int lane = t & 31;
    int s = src[e];
    int d = dst[e];
    float nrm = dinv[s] * dinv[d];
    float4 v = ((const float4*)(hw + (size_t)s * 128))[lane];
    float* o = agg + (size_t)d * 128 + lane * 4;
    atomicAdd(o + 0, v.x * nrm);
    atomicAdd(o + 1, v.y * nrm);
    atomicAdd(o + 2, v.z * nrm);
    atomicAdd(o + 3, v.w * nrm);
}

// h[i][j] = relu(agg[i][j] + hw[i][j]*dinv[i]^2 + b[j])   (self-loop folded in)
__global__ void node_epilogue_kernel(const float* __restrict__ agg, const float* __restrict__ hw,
                                     const float* __restrict__ dinv, const float* __restrict__ bias,
                                     float* __restrict__ hout, int n4) {
    int t = blockIdx.x * blockDim.x + threadIdx.x;
    if (t >= n4) return;
    int i = t >> 5;
    int q = t & 31;
    float dii = dinv[i]; dii *= dii;
    float4 a = ((const float4*)agg)[t];
    float4 w = ((const float4*)hw)[t];
    float4 b = ((const float4*)bias)[q];
    float4 r;
    r.x = fmaxf(a.x + w.x * dii + b.x, 0.0f);
    r.y = fmaxf(a.y + w.y * dii + b.y, 0.0f);
    r.z = fmaxf(a.z + w.z * dii + b.z, 0.0f);
    r.w = fmaxf(a.w + w.w * dii + b.w, 0.0f);
    ((float4*)hout)[t] = r;
}

// C[M,Nc] = relu?(A[M,K] @ B[K,Nc] + bias)  via V_WMMA_F32_16X16X4_F32.
// One wave per 16x16 tile. K multiple of 4; Nc multiple of 16.
// ISA f32 layouts: A 16x4: lanes0-15 M, {K0,K1}; lanes16-31 M, {K2,K3}.
//                  C/D 16x16: VGPR r -> M=r (lanes0-15) / M=r+8 (lanes16-31), N=lane%16.
__global__ void wmma_gemm_f32(const float* __restrict__ A, const float* __restrict__ B,
                              float* __restrict__ C, const float* __restrict__ bias,
                              int M, int K, int Nc, int relu) {
    int wave = (blockIdx.x * blockDim.x + threadIdx.x) >> 5;
    int lane = threadIdx.x & 31;
    int tilesN = Nc >> 4;
    int tilesM = (M + 15) >> 4;
    if (wave >= tilesM * tilesN) return;          // wave-uniform: EXEC stays all-1s
    int tm = wave / tilesN;
    int tn = wave - tm * tilesN;
    int row0 = tm << 4;
    int col0 = tn << 4;
    int mrow = lane & 15;
    int kb   = (lane >> 4) << 1;                  // 0 (lanes 0-15) or 2 (lanes 16-31)
    int arow = row0 + mrow; if (arow >= M) arow = M - 1;   // clamp (keeps EXEC uniform)
    int col  = col0 + mrow;

    v8f c;
    float bv = bias ? bias[col] : 0.0f;
    #pragma unroll
    for (int r = 0; r < 8; ++r) c[r] = bv;

    for (int k = 0; k < K; k += 4) {
        v2f a, b;
        a.x = A[(size_t)arow * K + k + kb];
        a.y = A[(size_t)arow * K + k + kb + 1];
        b.x = B[(size_t)(k + kb) * Nc + col];
        b.y = B[(size_t)(k + kb + 1) * Nc + col];
        c = __builtin_amdgcn_wmma_f32_16x16x4_f32(false, a, false, b, (short)0, c,
                                                  false, false);
    }

    int rbase = (lane >> 4) << 3;                 // lanes 16-31 hold M = r+8
    #pragma unroll
    for (int r = 0; r < 8; ++r) {
        int row = row0 + r + rbase;
        if (row < M) {
            float v = c[r];
            if (relu) v = fmaxf(v, 0.0f);
            C[(size_t)row * Nc + col] = v;
        }
    }
}

__global__ void count_kernel(const int* __restrict__ batch, float* __restrict__ cnts, int n) {
    int i = blockIdx.x * blockDim.x + threadIdx.x;
    if (i < n) atomicAdd(&cnts[batch[i]], 1.0f);
}

// One wave per node: sums[batch[i]] += h[i]
__global__ void pool_sum_kernel(const float* __restrict__ h, const int* __restrict__ batch,
                                float* __restrict__ sums, int N) {
    int t = blockIdx.x * blockDim.x + threadIdx.x;
    int i = t >> 5;
    if (i >= N) return;
    int lane = t & 31;
    int g = batch[i];
    float4 v = ((const float4*)(h + (size_t)i * 128))[lane];
    float* o = sums + (size_t)g * 128 + lane * 4;
    atomicAdd(o + 0, v.x);
    atomicAdd(o + 1, v.y);
    atomicAdd(o + 2, v.z);
    atomicAdd(o + 3, v.w);
}

__global__ void pool_div_kernel(const float* __restrict__ sums, const float* __restrict__ cnts,
                                float* __restrict__ pooled, int n4) {  // n4 = G*32
    int t = blockIdx.x * blockDim.x + threadIdx.x;
    if (t >= n4) return;
    int g = t >> 5;
    float inv = 1.0f / fmaxf(cnts[g], 1.0f);
    float4 s = ((const float4*)sums)[t];
    float4 r;
    r.x = s.x * inv; r.y = s.y * inv; r.z = s.z * inv; r.w = s.w * inv;
    ((float4*)pooled)[t] = r;
}

// One wave per graph: out[g] = fc1o[g,:] . w + b   (H2-wide dot, wave32 reduce)
__global__ void fc2_kernel(const float* __restrict__ fc1o, const float* __restrict__ w,
                           const float* __restrict__ b, float* __restrict__ out,
                           int G, int H2) {
    int t = blockIdx.x * blockDim.x + threadIdx.x;
    int g = t >> 5;
    if (g >= G) return;
    int lane = t & 31;
    float v = 0.0f;
    for (int j = lane; j < H2; j += 32) v += fc1o[(size_t)g * H2 + j] * w[j];
    for (int off = 16; off > 0; off >>= 1) v += __shfl_down(v, off, 32);
    if (lane == 0) out[g] = v + b[0];
}

static inline int cdiv(long long a, long long b) { return (int)((a + b - 1) / b); }

extern "C" void kernel_launch(void* const* d_in, const int* in_sizes, int n_in,
                              void* d_out, int out_size, void* d_ws, size_t ws_size,
                              hipStream_t stream) {
    const float* x     = (const float*)d_in[0];
    const int*   ei    = (const int*)  d_in[1];
    const int*   batch = (const int*)  d_in[2];
    const float* W0    = (const float*)d_in[3];
    const float* b0    = (const float*)d_in[4];
    const float* W1    = (const float*)d_in[5];
    const float* b1    = (const float*)d_in[6];
    const float* W2    = (const float*)d_in[7];
    const float* b2    = (const float*)d_in[8];
    const float* fc1w  = (const float*)d_in[9];
    const float* fc1b  = (const float*)d_in[10];
    const float* fc2w  = (const float*)d_in[11];
    const float* fc2b  = (const float*)d_in[12];
    float* out = (float*)d_out;

    const int N  = in_sizes[0];         // 100000
    const int E  = in_sizes[1] / 2;     // 1600000
    const int H  = in_sizes[4];         // 128
    const int H2 = in_sizes[10];        // 64
    const int G  = out_size;            // 512

    // Workspace carve-out (~155 MB): deg, dinv, h, hw, agg, sums, cnts, pooled, fc1o
    float* ws     = (float*)d_ws;
    float* deg    = ws;  ws += N;
    float* dinv   = ws;  ws += N;
    float* h      = ws;  ws += (size_t)N * H;
    float* hw     = ws;  ws += (size_t)N * H;
    float* agg    = ws;  ws += (size_t)N * H;
    float* sums   = ws;  ws += (size_t)G * H;
    float* cnts   = ws;  ws += G;
    float* pooled = ws;  ws += (size_t)G * H;
    float* fc1o   = ws;  ws += (size_t)G * H2;

    const int* src = ei;
    const int* dst = ei + E;
    const int BT = 256;
    const int n4 = N * (H / 4);         // N*32 float4 slots

    // --- degrees / normalization (re-initialized every call: replay-safe) ---
    fill_f32<<<cdiv(N, BT), BT, 0, stream>>>(deg, 1.0f, N);           // self loop
    degree_kernel<<<cdiv(E, BT), BT, 0, stream>>>(dst, deg, E);
    rsqrt_kernel<<<cdiv(N, BT), BT, 0, stream>>>(deg, dinv, N);

    // --- layer 0 pre-aggregation features: hw = x @ W0 (outer product) ---
    outer_kernel<<<cdiv(n4, BT), BT, 0, stream>>>(x, W0, hw, n4);

    const float* biases[3]  = { b0, b1, b2 };
    const float* nextW[3]   = { W1, W2, nullptr };
    const int gemmWaves = ((N + 15) / 16) * (H / 16);

    for (int l = 0; l < 3; ++l) {
        fill_f32<<<cdiv((size_t)N * H, BT), BT, 0, stream>>>(agg, 0.0f, N * H);
        edge_agg_kernel<<<cdiv((long long)E * 32, BT), BT, 0, stream>>>(
            src, dst, dinv, hw, agg, E);
        node_epilogue_kernel<<<cdiv(n4, BT), BT, 0, stream>>>(
            agg, hw, dinv, biases[l], h, n4);
        if (nextW[l]) {
            // hw = h @ W_{l+1}  (WMMA f32 16x16x4)
            wmma_gemm_f32<<<cdiv((long long)gemmWaves * 32, BT), BT, 0, stream>>>(
                h, nextW[l], hw, nullptr, N, H, H, 0);
        }
    }

    // --- global mean pool ---
    fill_f32<<<cdiv(G * H, BT), BT, 0, stream>>>(sums, 0.0f, G * H);
    fill_f32<<<cdiv(G, BT), BT, 0, stream>>>(cnts, 0.0f, G);
    count_kernel<<<cdiv(N, BT), BT, 0, stream>>>(batch, cnts, N);
    pool_sum_kernel<<<cdiv((long long)N * 32, BT), BT, 0, stream>>>(h, batch, sums, N);
    pool_div_kernel<<<cdiv(G * 32, BT), BT, 0, stream>>>(sums, cnts, pooled, G * 32);

    // --- MLP head: fc1 (WMMA + fused bias/ReLU), fc2 (wave dot) ---
    const int fc1Waves = (G / 16) * (H2 / 16);
    wmma_gemm_f32<<<cdiv((long long)fc1Waves * 32, BT), BT, 0, stream>>>(
        pooled, fc1w, fc1o, fc1b, G, H, H2, 1);
    fc2_kernel<<<cdiv((long long)G * 32, BT), BT, 0, stream>>>(fc1o, fc2w, fc2b, out, G, H2);
}